// GCN_16329465660164
// MI455X (gfx1250) — compile-verified
//
#include <hip/hip_runtime.h>
#include <hip/hip_bf16.h>

typedef __attribute__((ext_vector_type(2))) float v2f;
typedef __attribute__((ext_vector_type(8))) float v8f;

// ---------------------------------------------------------------------------
// Degree / normalization kernels
// ---------------------------------------------------------------------------
__global__ __launch_bounds__(256) void k_init_deg(float* __restrict__ deg, int n) {
  int i = blockIdx.x * blockDim.x + threadIdx.x;
  if (i < n) deg[i] = 1.0f;  // self-loop contributes 1 to every node's degree
}

__global__ __launch_bounds__(256) void k_count_deg(const int* __restrict__ col,
                                                   float* deg, int E) {
  int e = blockIdx.x * blockDim.x + threadIdx.x;
  if (e < E) {
    (void)__hip_atomic_fetch_add(&deg[col[e]], 1.0f, __ATOMIC_RELAXED,
                                 __HIP_MEMORY_SCOPE_AGENT);
  }
}

__global__ __launch_bounds__(256) void k_dinv(float* __restrict__ deg, int n) {
  int i = blockIdx.x * blockDim.x + threadIdx.x;
  if (i < n) {
    float d = deg[i];
    deg[i] = (d > 0.0f) ? rsqrtf(d) : 0.0f;  // in-place: deg -> deg^{-1/2}
  }
}

// ---------------------------------------------------------------------------
// f32 WMMA GEMM: C[M,N] = A[M,K] @ B[K,N]   (M%16==0, K%4==0, N%16==0)
// One wave32 computes one 16x16 C tile via V_WMMA_F32_16X16X4_F32.
//
// ISA 7.12.2 f32 fragment layouts (wave32):
//   A (16x4):  lanes 0-15 hold rows m0..m0+15, VGPR0=K0 VGPR1=K1;
//              lanes 16-31 same rows, VGPR0=K2 VGPR1=K3.
//   B (4x16):  lanes 0-15 hold cols n0..n0+15, VGPR0=K0 VGPR1=K1;
//              lanes 16-31 same cols, VGPR0=K2 VGPR1=K3.
//   C/D (16x16): VGPR v, lanes 0-15 -> row m0+v, lanes 16-31 -> row m0+8+v.
// ---------------------------------------------------------------------------
__global__ __launch_bounds__(256) void k_gemm_wmma_f32(
    const float* __restrict__ A, const float* __restrict__ B,
    float* __restrict__ C, int M, int K, int N) {
  const int lane  = threadIdx.x & 31;
  const int wave  = threadIdx.x >> 5;
  const int tilesN = N >> 4;
  const int t  = blockIdx.x * (blockDim.x >> 5) + wave;
  const int m0 = (t / tilesN) << 4;
  const int n0 = (t % tilesN) << 4;
  if (m0 >= M) return;

  const int l15   = lane & 15;
  const int khalf = (lane >> 4) << 1;  // 0 for lanes 0-15, 2 for lanes 16-31

  const float* arow = A + (size_t)(m0 + l15) * K + khalf;    // A[m][khalf..]
  const float* bcol = B + (size_t)khalf * N + (n0 + l15);    // B[khalf][n]

  v8f acc = {};
  for (int k0 = 0; k0 < K; k0 += 4) {
    v2f a, b;
    float2 av = *(const float2*)arow;        // A[m][k0+khalf], A[m][k0+khalf+1]
    a.x = av.x;
    a.y = av.y;
    b.x = bcol[0];                            // B[k0+khalf  ][n]
    b.y = bcol[N];                            // B[k0+khalf+1][n]
    acc = __builtin_amdgcn_wmma_f32_16x16x4_f32(
        /*neg_a=*/false, a, /*neg_b=*/false, b,
        /*c_mod=*/(short)0, acc, /*reuse_a=*/false, /*reuse_b=*/false);
    arow += 4;
    bcol += (size_t)4 * N;
  }

  const int rbase = m0 + ((lane >> 4) << 3);
  float* cptr = C + (size_t)rbase * N + n0 + l15;
#pragma unroll
  for (int v = 0; v < 8; ++v) cptr[(size_t)v * N] = acc[v];
}

// ---------------------------------------------------------------------------
// agg[i,:] = h[i,:] * dinv[i]^2    (self-loop term; also initializes agg)
// ---------------------------------------------------------------------------
__global__ __launch_bounds__(256) void k_selfloop(const float* __restrict__ h,
                                                  const float* __restrict__ dinv,
                                                  float* __restrict__ agg,
                                                  int total, int F) {
  int idx = blockIdx.x * blockDim.x + threadIdx.x;
  if (idx < total) {
    float di = dinv[idx / F];
    agg[idx] = h[idx] * di * di;
  }
}

// ---------------------------------------------------------------------------
// Edge scatter: agg[col[e],:] += h[row[e],:] * dinv[row]*dinv[col]
// One thread handles 4 features of one edge (float4 gather, 4 f32 atomics).
// ---------------------------------------------------------------------------
__global__ __launch_bounds__(256) void k_scatter(const float* __restrict__ h,
                                                 const int* __restrict__ row,
                                                 const int* __restrict__ col,
                                                 const float* __restrict__ dinv,
                                                 float* agg, int E, int F) {
  const int g = F >> 2;  // float4 groups per edge
  const long long total = (long long)E * g;
  long long tid = (long long)blockIdx.x * blockDim.x + threadIdx.x;
  if (tid >= total) return;
  const int e = (int)(tid / g);
  const int f = (int)(tid % g) << 2;
  const int r = row[e];
  const int c = col[e];
  const float nrm = dinv[r] * dinv[c];
  const float4 hv = *(const float4*)(h + (size_t)r * F + f);
  float* dst = agg + (size_t)c * F + f;
  (void)__hip_atomic_fetch_add(&dst[0], hv.x * nrm, __ATOMIC_RELAXED, __HIP_MEMORY_SCOPE_AGENT);
  (void)__hip_atomic_fetch_add(&dst[1], hv.y * nrm, __ATOMIC_RELAXED, __HIP_MEMORY_SCOPE_AGENT);
  (void)__hip_atomic_fetch_add(&dst[2], hv.z * nrm, __ATOMIC_RELAXED, __HIP_MEMORY_SCOPE_AGENT);
  (void)__hip_atomic_fetch_add(&dst[3], hv.w * nrm, __ATOMIC_RELAXED, __HIP_MEMORY_SCOPE_AGENT);
}

// ---------------------------------------------------------------------------
// agg[i,f] = (agg[i,f] + b[f])  [optionally ReLU], in place
// ---------------------------------------------------------------------------
__global__ __launch_bounds__(256) void k_bias_act(float* __restrict__ agg,
                                                  const float* __restrict__ b,
                                                  int total, int F, int do_relu) {
  int idx = blockIdx.x * blockDim.x + threadIdx.x;
  if (idx < total) {
    float v = agg[idx] + b[idx % F];
    if (do_relu) v = fmaxf(v, 0.0f);
    agg[idx] = v;
  }
}

// ---------------------------------------------------------------------------
extern "C" void kernel_launch(void* const* d_in, const int* in_sizes, int n_in,
                              void* d_out, int out_size, void* d_ws, size_t ws_size,
                              hipStream_t stream) {
  const float* x  = (const float*)d_in[0];
  const int*   ei = (const int*)d_in[1];   // [2, E] edge index (row; col)
  const float* W1 = (const float*)d_in[2];
  const float* b1 = (const float*)d_in[3];
  const float* W2 = (const float*)d_in[4];
  const float* b2 = (const float*)d_in[5];

  const int H  = in_sizes[3];         // 256
  const int C  = in_sizes[5];         // 64
  const int F  = in_sizes[2] / H;     // 128
  const int Nn = in_sizes[0] / F;     // 50000
  const int E  = in_sizes[1] / 2;     // 800000
  const int* rowIdx = ei;             // sources
  const int* colIdx = ei + E;         // targets

  // Workspace layout: dinv | hbuf (Nn*H) | abuf (Nn*H)
  auto al = [](size_t v) { return (v + 255) & ~(size_t)255; };
  float* dinv = (float*)d_ws;
  size_t off  = al((size_t)Nn * sizeof(float));
  float* hbuf = (float*)((char*)d_ws + off);
  off += al((size_t)Nn * H * sizeof(float));
  float* abuf = (float*)((char*)d_ws + off);

  const dim3 blk(256);
  auto blocks = [](long long work) { return (unsigned)((work + 255) / 256); };

  // --- symmetric degree normalization -------------------------------------
  k_init_deg <<<blocks(Nn), blk, 0, stream>>>(dinv, Nn);
  k_count_deg<<<blocks(E),  blk, 0, stream>>>(colIdx, dinv, E);
  k_dinv     <<<blocks(Nn), blk, 0, stream>>>(dinv, Nn);

  // --- layer 1: h = x @ W1; agg = D^-1/2 A D^-1/2 h; relu(agg + b1) -------
  const int tiles1 = ((Nn + 15) / 16) * (H / 16);
  k_gemm_wmma_f32<<<(tiles1 + 7) / 8, blk, 0, stream>>>(x, W1, hbuf, Nn, F, H);
  const int tot1 = Nn * H;
  k_selfloop<<<blocks(tot1), blk, 0, stream>>>(hbuf, dinv, abuf, tot1, H);
  k_scatter <<<blocks((long long)E * (H / 4)), blk, 0, stream>>>(hbuf, rowIdx, colIdx, dinv, abuf, E, H);
  k_bias_act<<<blocks(tot1), blk, 0, stream>>>(abuf, b1, tot1, H, /*relu=*/1);

  // --- layer 2: o = h1 @ W2; aggregate into d_out; add b2 -----------------
  const int tiles2 = ((Nn + 15) / 16) * (C / 16);
  k_gemm_wmma_f32<<<(tiles2 + 7) / 8, blk, 0, stream>>>(abuf, W2, hbuf, Nn, H, C);
  const int tot2 = Nn * C;
  float* out = (float*)d_out;
  k_selfloop<<<blocks(tot2), blk, 0, stream>>>(hbuf, dinv, out, tot2, C);
  k_scatter <<<blocks((long long)E * (C / 4)), blk, 0, stream>>>(hbuf, rowIdx, colIdx, dinv, out, E, C);
  k_bias_act<<<blocks(tot2), blk, 0, stream>>>(out, b2, tot2, C, /*relu=*/0);
}